// Voxelization_53317724013380
// MI455X (gfx1250) — compile-verified
//
#include <hip/hip_runtime.h>
#include <hip/hip_bf16.h>
#include <stdint.h>

// ---------------------------------------------------------------------------
// PointPillars voxelization for MI455X (gfx1250, wave32).
//   grid  = [432, 496, 1], BIG = 214272 voxel ids total
//   outputs: voxels [16000,32,4] f32 ; coords [16000,3] ; counts [16000]
// Strategy: occupancy bitmap over the (small) voxel-id space, wave32
// ballot/popc prefix scan to rank occupied ids (== sorted unique), atomic
// index-gather per kept voxel, per-voxel index sort to restore the
// reference's deterministic arrival order.
// ---------------------------------------------------------------------------

#define GXD 432
#define GYD 496
#define GZD 1
#define NVOX (GXD * GYD * GZD)        // 214272
#define MAXV 16000
#define MAXP 32
#define KCAP 64                        // gather cap per voxel (λ≈8.4 ⇒ safe)
#define SCAN_BLK 256
#define NBLK (NVOX / SCAN_BLK)         // 837 exactly

#define VOX_ELEMS  (MAXV * MAXP * 4)   // 2,048,000
#define COORD_ELEMS (MAXV * 3)

// workspace byte offsets
#define WS_OCC  0u          // NVOX*4 = 857088   (occupancy, then slotmap)
#define WS_BSUM 860160u     // NBLK*4
#define WS_BOFF 864256u     // NBLK*4
#define WS_UNIQ 868352u     // MAXV*4
#define WS_CNT  932352u     // MAXV*4
#define WS_IDX  1048576u    // MAXV*KCAP*4 = 4,096,000 -> ends at 5,144,576

// ---- gfx1250 async global->LDS path (guarded; falls back to plain loads) --
#if defined(__has_builtin)
#if __has_builtin(__builtin_amdgcn_global_load_async_to_lds_b128) && \
    __has_builtin(__builtin_amdgcn_s_wait_asynccnt)
#define USE_ASYNC_LDS 1
#endif
#endif
#ifndef USE_ASYNC_LDS
#define USE_ASYNC_LDS 0
#endif

#if USE_ASYNC_LDS
typedef int v4i_cdna5 __attribute__((ext_vector_type(4)));
typedef __attribute__((address_space(1))) v4i_cdna5 gv4i_cdna5;  // global
typedef __attribute__((address_space(3))) v4i_cdna5 lv4i_cdna5;  // LDS
#endif

// Voxel coordinate math — bit-exact mirror of the reference:
//   floor((p - lo) / vs), validity checked on the float results.
__device__ __forceinline__ int point_lin(float x, float y, float z) {
  float fx = floorf((x - 0.0f)   / 0.16f);
  float fy = floorf((y + 39.68f) / 0.16f);
  float fz = floorf((z + 3.0f)   / 4.0f);
  if (!(fx >= 0.0f && fx < 432.0f &&
        fy >= 0.0f && fy < 496.0f &&
        fz >= 0.0f && fz < 1.0f))
    return -1;
  return (int)fx * (GYD * GZD) + (int)fy * GZD + (int)fz;
}

// K0: re-initialize everything this launch depends on (poison-safe).
__global__ void k_init(float* __restrict__ vox, unsigned* __restrict__ occ,
                       unsigned* __restrict__ uniq, unsigned* __restrict__ cnt) {
  int i = blockIdx.x * 256 + threadIdx.x;
  if (i < NVOX) occ[i] = 0u;
  if (i < MAXV) { uniq[i] = (unsigned)NVOX; cnt[i] = 0u; }
  if (i < VOX_ELEMS) vox[i] = 0.0f;
}

// K1: mark occupied voxel ids. Streams 32 MB; optionally staged through LDS
// with the gfx1250 async-DMA path (ASYNCcnt-tracked global->LDS copy).
__global__ void k_mark(const float4* __restrict__ pts, unsigned* __restrict__ occ,
                       int n) {
  int i = blockIdx.x * 256 + threadIdx.x;
  float4 p;
#if USE_ASYNC_LDS
  __shared__ float4 stage[256];
  if (i < n) {
    __builtin_amdgcn_global_load_async_to_lds_b128(
        (gv4i_cdna5*)(pts + i),
        (lv4i_cdna5*)(&stage[threadIdx.x]),
        0, 0);
  }
  __builtin_amdgcn_s_wait_asynccnt(0);
  if (i >= n) return;
  p = stage[threadIdx.x];
#else
  if (i >= n) return;
  p = pts[i];
#endif
  int lin = point_lin(p.x, p.y, p.z);
  if (lin >= 0) occ[lin] = 1u;  // idempotent store; no atomic needed
}

// K2a: per-block occupancy sums via wave32 ballot.
__global__ void k_bsum(const unsigned* __restrict__ occ, unsigned* __restrict__ bsum) {
  int i = blockIdx.x * SCAN_BLK + threadIdx.x;  // grid covers NVOX exactly
  unsigned o = occ[i];
  unsigned long long m = __ballot(o != 0u);
  __shared__ unsigned wsum[SCAN_BLK / 32];
  int lane = threadIdx.x & 31, wid = threadIdx.x >> 5;
  if (lane == 0) wsum[wid] = (unsigned)__popcll(m);
  __syncthreads();
  if (threadIdx.x == 0) {
    unsigned s = 0;
    for (int w = 0; w < SCAN_BLK / 32; ++w) s += wsum[w];
    bsum[blockIdx.x] = s;
  }
}

// K2b: tiny serial exclusive scan over 837 block sums.
__global__ void k_scan1(const unsigned* __restrict__ bsum, unsigned* __restrict__ boff) {
  if (threadIdx.x == 0 && blockIdx.x == 0) {
    unsigned run = 0;
    for (int b = 0; b < NBLK; ++b) { boff[b] = run; run += bsum[b]; }
  }
}

// K2c: global rank per occupied id; emit uniq (first MAXV sorted occupied ids)
// and overwrite occ in place with slotmap (-1 = not a kept voxel).
__global__ void k_assign(unsigned* __restrict__ occ, const unsigned* __restrict__ boff,
                         unsigned* __restrict__ uniq) {
  int i = blockIdx.x * SCAN_BLK + threadIdx.x;
  unsigned o = occ[i];
  int lane = threadIdx.x & 31, wid = threadIdx.x >> 5;
  unsigned long long m = __ballot(o != 0u);
  unsigned lanepre = (unsigned)__popcll(m & ((1ull << lane) - 1ull));
  __shared__ unsigned wsum[SCAN_BLK / 32];
  if (lane == 0) wsum[wid] = (unsigned)__popcll(m);
  __syncthreads();
  unsigned base = boff[blockIdx.x];
  for (int w = 0; w < wid; ++w) base += wsum[w];
  unsigned rank = base + lanepre;
  int slot = -1;
  if (o && rank < (unsigned)MAXV) { slot = (int)rank; uniq[rank] = (unsigned)i; }
  occ[i] = (unsigned)slot;  // same-thread read-then-write: safe in place
}

// K3: count hits per kept voxel and gather point indices (cap KCAP).
__global__ void k_scatter(const float4* __restrict__ pts,
                          const unsigned* __restrict__ slotmap,
                          unsigned* __restrict__ cnt, unsigned* __restrict__ idxb,
                          int n) {
  int i = blockIdx.x * 256 + threadIdx.x;
  if (i >= n) return;
  float4 p = pts[i];
  int lin = point_lin(p.x, p.y, p.z);
  if (lin < 0) return;
  int s = (int)slotmap[lin];
  if (s < 0) return;
  unsigned pos = atomicAdd(&cnt[s], 1u);
  if (pos < (unsigned)KCAP) idxb[s * KCAP + pos] = (unsigned)i;
}

// K4: per voxel slot — sort gathered indices ascending (restores the
// reference's stable arrival order), write up to 32 points + coords + counts.
__global__ void k_write(const float4* __restrict__ pts,
                        const unsigned* __restrict__ uniq,
                        const unsigned* __restrict__ cnt,
                        const unsigned* __restrict__ idxb,
                        float4* __restrict__ vox,
                        float* __restrict__ coordsF, float* __restrict__ countsF) {
  int s = blockIdx.x * 256 + threadIdx.x;
  if (s >= MAXV) return;
  unsigned v = uniq[s];
  unsigned n = cnt[s];
  float c0 = -1.0f, c1 = -1.0f, c2 = -1.0f;
  int need = 0;
  if (v < (unsigned)NVOX) {
    unsigned cx = v / (GYD * GZD);
    unsigned rem = v % (GYD * GZD);
    c0 = (float)cx;
    c1 = (float)(rem / GZD);
    c2 = (float)(rem % GZD);
    need = (int)(n < (unsigned)MAXP ? n : (unsigned)MAXP);
  }
  coordsF[3 * s + 0] = c0;
  coordsF[3 * s + 1] = c1;
  coordsF[3 * s + 2] = c2;
  countsF[s] = (float)need;

  int m = (int)(n < (unsigned)KCAP ? n : (unsigned)KCAP);
  unsigned idxl[KCAP];
  for (int k = 0; k < m; ++k) {
    idxl[k] = idxb[s * KCAP + k];
    __builtin_prefetch(&pts[idxl[k]], 0, 0);  // warm L2 for the gather
  }
  // selection of the 'need' smallest original indices == reference ranks
  for (int r = 0; r < need; ++r) {
    int best = r;
    for (int k = r + 1; k < m; ++k)
      if (idxl[k] < idxl[best]) best = k;
    unsigned t = idxl[r]; idxl[r] = idxl[best]; idxl[best] = t;
    vox[s * MAXP + r] = pts[idxl[r]];
  }
}

extern "C" void kernel_launch(void* const* d_in, const int* in_sizes, int n_in,
                              void* d_out, int out_size, void* d_ws, size_t ws_size,
                              hipStream_t stream) {
  (void)n_in; (void)out_size; (void)ws_size;
  const float4* pts = (const float4*)d_in[0];
  int n = in_sizes[0] / 4;

  float* out = (float*)d_out;
  float* vox = out;                        // [16000,32,4]
  float* coordsF = out + VOX_ELEMS;        // [16000,3] as exact small floats
  float* countsF = coordsF + COORD_ELEMS;  // [16000]

  char* ws = (char*)d_ws;
  unsigned* occ  = (unsigned*)(ws + WS_OCC);   // occupancy -> slotmap
  unsigned* bsum = (unsigned*)(ws + WS_BSUM);
  unsigned* boff = (unsigned*)(ws + WS_BOFF);
  unsigned* uniq = (unsigned*)(ws + WS_UNIQ);
  unsigned* cnt  = (unsigned*)(ws + WS_CNT);
  unsigned* idxb = (unsigned*)(ws + WS_IDX);

  k_init<<<VOX_ELEMS / 256, 256, 0, stream>>>(vox, occ, uniq, cnt);
  if (n > 0)
    k_mark<<<(n + 255) / 256, 256, 0, stream>>>(pts, occ, n);
  k_bsum<<<NBLK, SCAN_BLK, 0, stream>>>(occ, bsum);
  k_scan1<<<1, 32, 0, stream>>>(bsum, boff);
  k_assign<<<NBLK, SCAN_BLK, 0, stream>>>(occ, boff, uniq);
  if (n > 0)
    k_scatter<<<(n + 255) / 256, 256, 0, stream>>>(pts, occ, cnt, idxb, n);
  k_write<<<(MAXV + 255) / 256, 256, 0, stream>>>(pts, uniq, cnt, idxb,
                                                  (float4*)vox, coordsF, countsF);
}